// CRFOutputLayer_14834817040952
// MI455X (gfx1250) — compile-verified
//
#include <hip/hip_runtime.h>

// CRF output layer for MI455X (gfx1250, wave32).
//   Phase 1: emissions = outputs @ fc_w^T + fc_b   -- bandwidth-bound fp32 GEMM
//            using V_WMMA_F32_16X16X4_F32; A tiles staged to LDS via
//            GLOBAL_LOAD_ASYNC_TO_LDS_B128 (ASYNCcnt-tracked direct DMA).
//   Phase 2: per-batch CRF log-likelihood (numerator + forward-algorithm denom),
//            one wave32 per batch, lane = label, score broadcast via __shfl.
//   Phase 3: loss = -mean(llh).

#define HIDDEN  768
#define NLAB    17
#define BB      64
#define TT      512
#define LDSW    (HIDDEN + 4)   // 772: 772 % 64 == 4 -> 16 rows hit distinct LDS banks

typedef __attribute__((ext_vector_type(2))) float v2f;
typedef __attribute__((ext_vector_type(4))) float v4f;
typedef __attribute__((ext_vector_type(8))) float v8f;
typedef __attribute__((ext_vector_type(4))) int   i4;

#if __has_builtin(__builtin_amdgcn_global_load_async_to_lds_b128)
#define HAS_ASYNC_LDS 1
typedef __attribute__((address_space(1))) i4 gi4;   // global int4
typedef __attribute__((address_space(3))) i4 li4;   // LDS int4
#else
#define HAS_ASYNC_LDS 0
#endif

// ---------------------------------------------------------------------------
// Phase 1: emissions GEMM. One wave per 16-row M tile (2048 blocks x 32 thr).
// ---------------------------------------------------------------------------
__global__ __launch_bounds__(32) void emis_gemm_kernel(
    const float* __restrict__ outputs,   // [32768, 768]
    const float* __restrict__ fc_w,      // [17, 768]
    const float* __restrict__ fc_b,      // [17]
    float* __restrict__ emis)            // [32768, 17]
{
    __shared__ float lds[16 * LDSW];     // 49,408 B: padded 16x768 fp32 A tile

    const int  lane = threadIdx.x;       // wave32
    const int  half = lane >> 4;         // 0: K+{0,1}, 1: K+{2,3} (A/B layout)
    const int  l15  = lane & 15;
    const long m0   = (long)blockIdx.x * 16;

    // ---- Stage A tile into LDS (coalesced 128-bit per lane) ----
#if HAS_ASYNC_LDS
    // Async DMA path: global -> LDS directly, no VGPR round trip (ASYNCcnt).
    for (int row = 0; row < 16; ++row) {
        const float* src = outputs + (m0 + row) * (long)HIDDEN;
        for (int col = lane * 4; col < HIDDEN; col += 32 * 4) {
            __builtin_amdgcn_global_load_async_to_lds_b128(
                (gi4*)(src + col), (li4*)&lds[row * LDSW + col], 0, 0);
        }
    }
#if __has_builtin(__builtin_amdgcn_s_wait_asynccnt)
    __builtin_amdgcn_s_wait_asynccnt(0);
#else
    asm volatile("s_wait_asynccnt 0" ::: "memory");
#endif
#else
    // Fallback: non-temporal b128 loads (outputs is read exactly once).
    for (int row = 0; row < 16; ++row) {
        const float* src = outputs + (m0 + row) * (long)HIDDEN;
        for (int col = lane * 4; col < HIDDEN; col += 32 * 4) {
            v4f v = __builtin_nontemporal_load((const v4f*)(src + col));
            *(v4f*)(&lds[row * LDSW + col]) = v;
        }
    }
#endif
    __syncthreads();

    // A operand (16x4 f32): lanes 0-15 hold row l15 K={k,k+1}; lanes 16-31 K={k+2,k+3}.
    // B operand (4x16 f32): B[k][n] = fc_w[n][k]; same contiguous-pair pattern.
    v8f acc0 = {};   // labels 0..15
    v8f acc1 = {};   // labels 16..31 (only 16 is real; B masked to zero elsewhere)

    const bool n16 = (l15 == 0);         // lane owns the real column of tile 1
    for (int k = 0; k < HIDDEN; k += 4) {
        const int kk = k + 2 * half;

        v2f a   = *(const v2f*)(&lds[l15 * LDSW + kk]);      // ds_load_b64
        v2f b0  = *(const v2f*)(fc_w + l15 * HIDDEN + kk);   // L1-resident
        // Broadcast load (same address wave-wide per half) + data-select:
        // avoids EXEC-mask save/restore in the WMMA loop.
        v2f b1w = *(const v2f*)(fc_w + 16 * HIDDEN + kk);
        v2f b1;
        b1.x = n16 ? b1w.x : 0.0f;
        b1.y = n16 ? b1w.y : 0.0f;

        acc0 = __builtin_amdgcn_wmma_f32_16x16x4_f32(
                   false, a, false, b0, (short)0, acc0, false, false);
        acc1 = __builtin_amdgcn_wmma_f32_16x16x4_f32(
                   false, a, false, b1, (short)0, acc1, false, false);
    }

    // C/D layout: VGPR r -> (M=r, N=lane) lanes 0-15; (M=r+8, N=lane-16) lanes 16-31.
    const float bias0  = fc_b[l15];
    const float bias16 = fc_b[16];
    for (int r = 0; r < 8; ++r) {
        const int M  = r + 8 * half;
        float* dst = emis + (m0 + M) * (long)NLAB;
        dst[l15] = acc0[r] + bias0;
        if (n16) dst[16] = acc1[r] + bias16;
    }
}

// ---------------------------------------------------------------------------
// Phase 2: per-batch CRF log-likelihood. One wave32 per batch; lane = label j.
// ---------------------------------------------------------------------------
__global__ __launch_bounds__(32) void crf_llh_kernel(
    const float* __restrict__ emis,
    const int* __restrict__ labels,
    const unsigned char* __restrict__ mask,
    const float* __restrict__ start_trans,
    const float* __restrict__ end_trans,
    const float* __restrict__ trans,
    float* __restrict__ llh)             // [64] workspace
{
    const int b    = blockIdx.x;
    const int lane = threadIdx.x;
    const float*         em = emis   + (long)b * TT * NLAB;
    const int*           tg = labels + b * TT;
    const unsigned char* mk = mask   + b * TT;

    // ---- numerator: lanes stride over t, then warp-reduce ----
    float num = 0.0f;
    int   cnt = 0;                       // sum(mask) with mask[0] forced true
    for (int t = lane; t < TT; t += 32) {
        int tag = tg[t]; if (tag == -100) tag = 0;
        const bool m = (t == 0) ? true : (mk[t] != 0);
        cnt += m ? 1 : 0;
        if (t == 0) {
            num += start_trans[tag] + em[tag];
        } else if (m) {
            int tp = tg[t - 1]; if (tp == -100) tp = 0;
            num += trans[tp * NLAB + tag] + em[t * NLAB + tag];
        }
    }
    for (int off = 16; off > 0; off >>= 1) {
        num += __shfl_down(num, off, 32);
        cnt += __shfl_down(cnt, off, 32);
    }
    num = __shfl(num, 0, 32);
    cnt = __shfl(cnt, 0, 32);

    // ---- denominator: forward algorithm, lane j holds score[j] ----
    const int  j   = lane;
    const bool act = (j < NLAB);
    float tr[NLAB];                      // tr[i] = trans[i][j]
    {
        const int jc = act ? j : (NLAB - 1);
        for (int i = 0; i < NLAB; ++i) tr[i] = trans[i * NLAB + jc];
    }
    float score = act ? (start_trans[j] + em[j]) : -3.0e38f;

    for (int t = 1; t < TT; ++t) {
        const float emj = act ? em[t * NLAB + j] : 0.0f;
        float vv[NLAB];
        float mx = -3.0e38f;
        for (int i = 0; i < NLAB; ++i) {          // broadcast score[i] wave-wide
            const float si = __shfl(score, i, 32);
            const float v  = si + tr[i];
            vv[i] = v;
            mx = fmaxf(mx, v);
        }
        float s = 0.0f;
        for (int i = 0; i < NLAB; ++i) s += __expf(vv[i] - mx);
        const float nxt = emj + mx + __logf(s);
        score = (mk[t] != 0) ? nxt : score;
    }

    // logsumexp over lanes 0..16 of score + end_trans[j]
    const float v = act ? (score + end_trans[j]) : -3.0e38f;
    float mx = v;
    for (int off = 16; off > 0; off >>= 1) mx = fmaxf(mx, __shfl_xor(mx, off, 32));
    float se = act ? __expf(v - mx) : 0.0f;
    for (int off = 16; off > 0; off >>= 1) se += __shfl_xor(se, off, 32);

    if (lane == 0) {
        const float denom = mx + __logf(se);
        int last = tg[cnt - 1]; if (last == -100) last = 0;
        llh[b] = (num + end_trans[last]) - denom;
    }
}

// ---------------------------------------------------------------------------
// Phase 3: loss = -mean(llh). Overwrites (never accumulates) -> replay-safe.
// ---------------------------------------------------------------------------
__global__ __launch_bounds__(64) void loss_kernel(
    const float* __restrict__ llh, float* __restrict__ loss_out)
{
    __shared__ float part[2];
    const int t = threadIdx.x;
    float v = llh[t];
    for (int off = 16; off > 0; off >>= 1) v += __shfl_down(v, off, 32);
    if ((t & 31) == 0) part[t >> 5] = v;
    __syncthreads();
    if (t == 0) loss_out[0] = -(part[0] + part[1]) * (1.0f / (float)BB);
}

// ---------------------------------------------------------------------------
extern "C" void kernel_launch(void* const* d_in, const int* in_sizes, int n_in,
                              void* d_out, int out_size, void* d_ws, size_t ws_size,
                              hipStream_t stream)
{
    const float*         outputs     = (const float*)d_in[0];
    const float*         fc_w        = (const float*)d_in[1];
    const float*         fc_b        = (const float*)d_in[2];
    const float*         start_trans = (const float*)d_in[3];
    const float*         end_trans   = (const float*)d_in[4];
    const float*         trans       = (const float*)d_in[5];
    const int*           labels      = (const int*)d_in[6];
    const unsigned char* mask        = (const unsigned char*)d_in[7];

    float* emis = (float*)d_out;                          // [64*512*17]
    float* loss = emis + (size_t)BB * TT * NLAB;          // d_out tail
    float* llh  = (float*)d_ws;                           // [64] scratch

    emis_gemm_kernel<<<(BB * TT) / 16, 32, 0, stream>>>(outputs, fc_w, fc_b, emis);
    crf_llh_kernel<<<BB, 32, 0, stream>>>(emis, labels, mask,
                                          start_trans, end_trans, trans, llh);
    loss_kernel<<<1, 64, 0, stream>>>(llh, loss);
}